// TransformerBlock_82944408420967
// MI455X (gfx1250) — compile-verified
//
#include <hip/hip_runtime.h>

typedef __attribute__((ext_vector_type(2))) float v2f;
typedef __attribute__((ext_vector_type(4))) float v4f;
typedef __attribute__((ext_vector_type(8))) float v8f;

#define B_      8
#define D_      4096
#define H_      32
#define HD_     128
#define MAXSEQ_ 2048
#define AL_     10
#define FF_     11008
#define EPS_    1e-5f
#define SCALE_  0.08838834764831845f   // 1/sqrt(128)

// ---------------- helpers ----------------
__device__ __forceinline__ float wave_sum(float v) {
#pragma unroll
  for (int off = 16; off > 0; off >>= 1) v += __shfl_xor(v, off, 32);
  return v;
}
__device__ __forceinline__ float wave_max(float v) {
#pragma unroll
  for (int off = 16; off > 0; off >>= 1) v = fmaxf(v, __shfl_xor(v, off, 32));
  return v;
}

// ---------------- tiny elementwise kernels ----------------
__global__ void fill0_k(float* __restrict__ p, int n) {
  int i = blockIdx.x * 256 + threadIdx.x;
  if (i < n) p[i] = 0.f;
}
__global__ void copy_k(float* __restrict__ d, const float* __restrict__ s, int n) {
  int i = blockIdx.x * 256 + threadIdx.x;
  if (i < n) d[i] = s[i];
}
__global__ void silu_mul_k(const float* __restrict__ g, const float* __restrict__ u,
                           float* __restrict__ s, int n) {
  int i = blockIdx.x * 256 + threadIdx.x;
  if (i < n) {
    float x = g[i];
    s[i] = (x / (1.f + __expf(-x))) * u[i];
  }
}

// ---------------- RMSNorm: one block per row ----------------
__global__ void rmsnorm_k(const float* __restrict__ x, const float* __restrict__ w,
                          float* __restrict__ o) {
  const int rowbase = blockIdx.x * D_;
  __shared__ float red[8];
  __shared__ float bscale;
  float s = 0.f;
  for (int i = threadIdx.x; i < D_; i += 256) {
    float v = x[rowbase + i];
    s += v * v;
  }
  s = wave_sum(s);
  const int lane = threadIdx.x & 31, w8 = threadIdx.x >> 5;
  if (lane == 0) red[w8] = s;
  __syncthreads();
  if (threadIdx.x == 0) {
    float t = 0.f;
#pragma unroll
    for (int i = 0; i < 8; ++i) t += red[i];
    bscale = rsqrtf(t / (float)D_ + EPS_);
  }
  __syncthreads();
  const float sc = bscale;
  for (int i = threadIdx.x; i < D_; i += 256)
    o[rowbase + i] = x[rowbase + i] * sc * w[i];
}

// ---------------- RoPE on xq and xk (interleaved pairs) ----------------
__global__ void rope_k(float* __restrict__ xq, float* __restrict__ xk,
                       const float* __restrict__ fc, const float* __restrict__ fs) {
  int t = blockIdx.x * 256 + threadIdx.x;
  if (t >= B_ * H_ * (HD_ / 2)) return;
  const int i = t & 63;          // freq index
  const int bh = t >> 6;         // flattened (b,h)
  const int base = bh * HD_ + 2 * i;
  const float c = fc[i], s = fs[i];
  float a = xq[base], b = xq[base + 1];
  xq[base] = a * c - b * s;
  xq[base + 1] = a * s + b * c;
  a = xk[base]; b = xk[base + 1];
  xk[base] = a * c - b * s;
  xk[base + 1] = a * s + b * c;
}

// ---------------- split-K WMMA f32 GEMM: C[16 x N] += A[16 x K] * B[K x N] ----
// One wave per block. grid = (N/16, KSPLIT). Rows >= Mvalid read from a
// zero-filled row (zrow) so all loads are unconditional (no EXEC juggling in
// the hot loop). B and A addresses are strength-reduced pointer increments.
// A lane layout (16x4 f32 A): lane&15 = M, lane>>4 selects K pair {0,1}/{2,3}.
// C/D layout: 8 VGPRs; lane<16 -> M=r, lane>=16 -> M=r+8; N = lane&15 + n0.
__global__ void gemm16_wmma(const float* __restrict__ A, int Mvalid, int K,
                            const float* __restrict__ B, int N,
                            float* __restrict__ C, const float* __restrict__ zrow) {
  const int tid  = threadIdx.x;
  const int ln15 = tid & 15;       // A row / B col / C col
  const int kh   = tid >> 4;       // K-pair half and C row-half
  const int n0   = blockIdx.x << 4;
  const int kchunk = K / gridDim.y;
  const int kbeg = blockIdx.y * kchunk;

  const float* __restrict__ Ap =
      ((ln15 < Mvalid) ? (A + (size_t)ln15 * K) : zrow) + kbeg + 2 * kh;
  const float* __restrict__ Bp =
      B + (size_t)(kbeg + 2 * kh) * N + n0 + ln15;
  const size_t strideB = (size_t)4 * N;

  v8f acc = {0.f, 0.f, 0.f, 0.f, 0.f, 0.f, 0.f, 0.f};
#pragma unroll 8
  for (int k = 0; k < kchunk; k += 4) {
    const v2f a = *(const v2f*)Ap;                      // one b64 load
    v2f b;
    b.x = __builtin_nontemporal_load(Bp);               // streamed weights: NT
    b.y = __builtin_nontemporal_load(Bp + N);
    __builtin_prefetch(Bp + 16 * (size_t)N, 0, 1);      // global_prefetch_b8
    acc = __builtin_amdgcn_wmma_f32_16x16x4_f32(
        /*neg_a=*/false, a, /*neg_b=*/false, b,
        /*c_mod=*/(short)0, acc, /*reuse_a=*/false, /*reuse_b=*/false);
    Ap += 4;
    Bp += strideB;
  }
#pragma unroll
  for (int r = 0; r < 8; ++r) {
    const int m = r + (kh << 3);
    if (m < Mvalid) {
      __hip_atomic_fetch_add(&C[(size_t)m * N + n0 + ln15], acc[r],
                             __ATOMIC_RELAXED, __HIP_MEMORY_SCOPE_AGENT);
    }
  }
}

// ---------------- flash-decode attention + fused adapter attention ----------
// grid (B, H), 256 threads (8 waves). Wave-per-key: each lane holds 4 q dims
// (coalesced v4f per key row), shuffle-reduced dot products. KV cache rows are
// nontemporal (537 MB stream > 192 MB L2). Key/value index 2047 substitutes
// the freshly computed xk/xv (functional cache update; inputs never mutated).
__global__ void attn_k(const float* __restrict__ xq, const float* __restrict__ xk,
                       const float* __restrict__ xv,
                       const float* __restrict__ cache_k, const float* __restrict__ cache_v,
                       const float* __restrict__ ak, const float* __restrict__ av,
                       const float* __restrict__ gate, float* __restrict__ attn_out) {
  const int b = blockIdx.x, h = blockIdx.y;
  const int tid = threadIdx.x;
  const int lane = tid & 31, wv8 = tid >> 5;

  __shared__ float sc[MAXSEQ_];
  __shared__ float red[8];
  __shared__ float ow[8][HD_];
  __shared__ float o_final[HD_];
  __shared__ float asc[16];
  __shared__ float bc;

  const size_t qoff = (size_t)b * D_ + (size_t)h * HD_;
  const v4f q4 = *(const v4f*)(xq + qoff + lane * 4);
  const size_t kbase = (size_t)b * MAXSEQ_ * D_ + (size_t)h * HD_;

  // ---- scores ----
  for (int j = wv8; j < MAXSEQ_; j += 8) {
    v4f k4;
    if (j == MAXSEQ_ - 1) {
      k4 = *(const v4f*)(xk + qoff + lane * 4);
    } else {
      k4 = __builtin_nontemporal_load(
          (const v4f*)(cache_k + kbase + (size_t)j * D_ + lane * 4));
    }
    float d = q4.x * k4.x + q4.y * k4.y + q4.z * k4.z + q4.w * k4.w;
    d = wave_sum(d);
    if (lane == 0) sc[j] = d * SCALE_;
  }
  __syncthreads();

  // ---- softmax ----
  float m = -1e30f;
  for (int j = tid; j < MAXSEQ_; j += 256) m = fmaxf(m, sc[j]);
  m = wave_max(m);
  if (lane == 0) red[wv8] = m;
  __syncthreads();
  if (tid == 0) {
    float mm = red[0];
#pragma unroll
    for (int w = 1; w < 8; ++w) mm = fmaxf(mm, red[w]);
    bc = mm;
  }
  __syncthreads();
  m = bc;
  float s = 0.f;
  for (int j = tid; j < MAXSEQ_; j += 256) {
    float e = __expf(sc[j] - m);
    sc[j] = e;
    s += e;
  }
  s = wave_sum(s);
  if (lane == 0) red[wv8] = s;
  __syncthreads();
  if (tid == 0) {
    float t = 0.f;
#pragma unroll
    for (int w = 0; w < 8; ++w) t += red[w];
    bc = 1.f / t;
  }
  __syncthreads();
  const float inv = bc;

  // ---- P @ V ----
  v4f o = {0.f, 0.f, 0.f, 0.f};
  for (int j = wv8; j < MAXSEQ_; j += 8) {
    v4f v4;
    if (j == MAXSEQ_ - 1) {
      v4 = *(const v4f*)(xv + qoff + lane * 4);
    } else {
      v4 = __builtin_nontemporal_load(
          (const v4f*)(cache_v + kbase + (size_t)j * D_ + lane * 4));
    }
    const float p = sc[j];
    o.x += p * v4.x; o.y += p * v4.y; o.z += p * v4.z; o.w += p * v4.w;
  }
  ow[wv8][lane * 4 + 0] = o.x;
  ow[wv8][lane * 4 + 1] = o.y;
  ow[wv8][lane * 4 + 2] = o.z;
  ow[wv8][lane * 4 + 3] = o.w;
  __syncthreads();
  if (tid < HD_) {
    float t = 0.f;
#pragma unroll
    for (int w = 0; w < 8; ++w) t += ow[w][tid];
    o_final[tid] = t * inv;
  }

  // ---- adapter attention (separate softmax over AL=10) ----
  for (int k = wv8; k < AL_; k += 8) {
    const float* ap = ak + (size_t)k * D_ + (size_t)h * HD_;
    const v4f a4 = *(const v4f*)(ap + lane * 4);
    float d = q4.x * a4.x + q4.y * a4.y + q4.z * a4.z + q4.w * a4.w;
    d = wave_sum(d);
    if (lane == 0) asc[k] = d * SCALE_;
  }
  __syncthreads();
  if (tid == 0) {
    float mm = asc[0];
    for (int k = 1; k < AL_; ++k) mm = fmaxf(mm, asc[k]);
    float t = 0.f;
    for (int k = 0; k < AL_; ++k) {
      float e = __expf(asc[k] - mm);
      asc[k] = e;
      t += e;
    }
    float it = 1.f / t;
    for (int k = 0; k < AL_; ++k) asc[k] *= it;
  }
  __syncthreads();
  if (tid < HD_) {
    float ao = 0.f;
#pragma unroll
    for (int k = 0; k < AL_; ++k)
      ao += asc[k] * av[(size_t)k * D_ + (size_t)h * HD_ + tid];
    const float g = tanhf(gate[h]);
    attn_out[qoff + tid] = o_final[tid] + g * ao;
  }
}

// ---------------- host-side orchestration ----------------
extern "C" void kernel_launch(void* const* d_in, const int* in_sizes, int n_in,
                              void* d_out, int out_size, void* d_ws, size_t ws_size,
                              hipStream_t stream) {
  (void)in_sizes; (void)n_in; (void)out_size; (void)ws_size;
  const float* x       = (const float*)d_in[0];
  const float* adapter = (const float*)d_in[1];
  const float* cache_k = (const float*)d_in[2];
  const float* cache_v = (const float*)d_in[3];
  const float* fcos    = (const float*)d_in[4];
  const float* fsin    = (const float*)d_in[5];
  const float* wq      = (const float*)d_in[6];
  const float* wk      = (const float*)d_in[7];
  const float* wv      = (const float*)d_in[8];
  const float* wo      = (const float*)d_in[9];
  const float* gate    = (const float*)d_in[10];
  const float* anw     = (const float*)d_in[11];
  const float* fnw     = (const float*)d_in[12];
  const float* w1      = (const float*)d_in[13];
  const float* w2      = (const float*)d_in[14];
  const float* w3      = (const float*)d_in[15];
  float* out = (float*)d_out;

  // workspace layout (floats)
  float* ws   = (float*)d_ws;
  float* xn   = ws;                  // 8*4096
  float* xq   = xn  + B_ * D_;
  float* xk   = xq  + B_ * D_;
  float* xv   = xk  + B_ * D_;
  float* akb  = xv  + B_ * D_;       // 10*4096
  float* avb  = akb + AL_ * D_;
  float* attn = avb + AL_ * D_;      // 8*4096
  float* h    = attn + B_ * D_;
  float* hn   = h   + B_ * D_;
  float* g    = hn  + B_ * D_;       // 8*11008
  float* u    = g   + B_ * FF_;
  float* sg   = u   + B_ * FF_;
  float* zrow = sg  + B_ * FF_;      // 11012 zeros (padded-A row)

  const int nBD = B_ * D_;     // 32768
  const int nAD = AL_ * D_;    // 40960
  const int nBF = B_ * FF_;    // 88064
  const int nZ  = FF_ + 4;     // zero row covers max K

  // 0) zero row for padded-A lanes
  fill0_k<<<(nZ + 255) / 256, 256, 0, stream>>>(zrow, nZ);

  // 1) attention RMSNorm
  rmsnorm_k<<<B_, 256, 0, stream>>>(x, anw, xn);

  // 2) zero split-K accumulators, then QKV + adapter-KV GEMMs (WMMA f32)
  fill0_k<<<(3 * nBD + 255) / 256, 256, 0, stream>>>(xq, 3 * nBD);
  fill0_k<<<(2 * nAD + 255) / 256, 256, 0, stream>>>(akb, 2 * nAD);
  gemm16_wmma<<<dim3(D_ / 16, 16), 32, 0, stream>>>(xn, B_, D_, wq, D_, xq, zrow);
  gemm16_wmma<<<dim3(D_ / 16, 16), 32, 0, stream>>>(xn, B_, D_, wk, D_, xk, zrow);
  gemm16_wmma<<<dim3(D_ / 16, 16), 32, 0, stream>>>(xn, B_, D_, wv, D_, xv, zrow);
  gemm16_wmma<<<dim3(D_ / 16, 16), 32, 0, stream>>>(adapter, AL_, D_, wk, D_, akb, zrow);
  gemm16_wmma<<<dim3(D_ / 16, 16), 32, 0, stream>>>(adapter, AL_, D_, wv, D_, avb, zrow);

  // 3) RoPE on q,k
  rope_k<<<(B_ * H_ * (HD_ / 2) + 255) / 256, 256, 0, stream>>>(xq, xk, fcos, fsin);

  // 4) flash-decode attention (+ fused adapter attention, gate)
  attn_k<<<dim3(B_, H_), 256, 0, stream>>>(xq, xk, xv, cache_k, cache_v,
                                           akb, avb, gate, attn);

  // 5) h = x + attn @ wo   (residual pre-loaded, GEMM accumulates atomically)
  copy_k<<<(nBD + 255) / 256, 256, 0, stream>>>(h, x, nBD);
  gemm16_wmma<<<dim3(D_ / 16, 16), 32, 0, stream>>>(attn, B_, D_, wo, D_, h, zrow);

  // 6) FFN RMSNorm
  rmsnorm_k<<<B_, 256, 0, stream>>>(h, fnw, hn);

  // 7) g = hn@w1, u = hn@w3, sg = silu(g)*u
  fill0_k<<<(2 * nBF + 255) / 256, 256, 0, stream>>>(g, 2 * nBF);
  gemm16_wmma<<<dim3(FF_ / 16, 8), 32, 0, stream>>>(hn, B_, D_, w1, FF_, g, zrow);
  gemm16_wmma<<<dim3(FF_ / 16, 8), 32, 0, stream>>>(hn, B_, D_, w3, FF_, u, zrow);
  silu_mul_k<<<(nBF + 255) / 256, 256, 0, stream>>>(g, u, sg, nBF);

  // 8) out = h + sg @ w2   (residual pre-loaded into d_out, accumulate)
  copy_k<<<(nBD + 255) / 256, 256, 0, stream>>>(out, h, nBD);
  gemm16_wmma<<<dim3(D_ / 16, 16), 32, 0, stream>>>(sg, B_, FF_, w2, D_, out, zrow);
}